// IndexedLinear_80401787781505
// MI455X (gfx1250) — compile-verified
//
#include <hip/hip_runtime.h>
#include <hip/hip_bf16.h>
#include <stddef.h>

// ---------------------------------------------------------------------------
// IndexedLinear on MI455X (gfx1250):
//   out[n] = x[n] @ W[idx[n]],  N=262144, D_IN=D_OUT=256, 16 types, fp32.
//
// v3: same structure as v2 (type-binned rows, 64-row supertiles, packed-W
// b64 B fragments, fp32 V_WMMA_F32_16X16X4_F32 core), but the store epilogue
// is branchless: pad rows redirect to a dummy scratch row instead of a
// per-element exec-mask branch.
// ---------------------------------------------------------------------------

typedef __attribute__((ext_vector_type(2))) float v2f;
typedef __attribute__((ext_vector_type(8))) float v8f;

#define DIN 256
#define DOUT 256
#define NTYPES 16
#define ROWS_PER_BLOCK 64      // 4 sub-tiles of 16 rows
#define LDSW 260               // A row stride (floats): 260 % 64 == 4 -> conflict-free b64

// ---------------- prologue kernels (binning) ----------------

__global__ void il_init_kernel(int* counts) {
    if (threadIdx.x < NTYPES) counts[threadIdx.x] = 0;
}

__global__ void il_hist_kernel(const int* __restrict__ idx, int* __restrict__ counts, int n) {
    __shared__ int h[NTYPES];
    if (threadIdx.x < NTYPES) h[threadIdx.x] = 0;
    __syncthreads();
    int stride = gridDim.x * blockDim.x;
    for (int i = blockIdx.x * blockDim.x + threadIdx.x; i < n; i += stride)
        atomicAdd(&h[idx[i] & (NTYPES - 1)], 1);
    __syncthreads();
    if (threadIdx.x < NTYPES) atomicAdd(&counts[threadIdx.x], h[threadIdx.x]);
}

// Per-type segment offsets aligned to 64 rows; pad slots = -1; per-supertile
// descriptors (type + row base).
__global__ void il_setup_kernel(const int* __restrict__ counts,
                                int* __restrict__ cursors,
                                int* __restrict__ tileType,
                                int* __restrict__ tileRowBase,
                                int* __restrict__ rowlist,
                                int maxTiles) {
    __shared__ int s_off[NTYPES];
    __shared__ int s_tileStart[NTYPES + 1];
    if (threadIdx.x == 0) {
        int off = 0, ts = 0;
        for (int t = 0; t < NTYPES; ++t) {
            int c  = counts[t];
            int ca = (c + (ROWS_PER_BLOCK - 1)) & ~(ROWS_PER_BLOCK - 1);
            s_off[t]       = off;
            cursors[t]     = off;
            s_tileStart[t] = ts;
            for (int p = c; p < ca; ++p) rowlist[off + p] = -1;   // pad rows
            off += ca;
            ts  += ca / ROWS_PER_BLOCK;
        }
        s_tileStart[NTYPES] = ts;
    }
    __syncthreads();
    int nt = s_tileStart[NTYPES];
    for (int j = threadIdx.x; j < maxTiles; j += blockDim.x) {
        int ty = -1, rb = 0;
        if (j < nt) {
            for (int t = 0; t < NTYPES; ++t) {
                if (j >= s_tileStart[t] && j < s_tileStart[t + 1]) {
                    ty = t;
                    rb = s_off[t] + (j - s_tileStart[t]) * ROWS_PER_BLOCK;
                }
            }
        }
        tileType[j]    = ty;
        tileRowBase[j] = rb;
    }
}

// Block-local compaction scatter (1 global atomic per type per block).
__global__ void il_scatter_kernel(const int* __restrict__ idx,
                                  int* __restrict__ cursors,
                                  int* __restrict__ rowlist, int n) {
    __shared__ int h[NTYPES];
    __shared__ int base[NTYPES];
    if (threadIdx.x < NTYPES) h[threadIdx.x] = 0;
    __syncthreads();
    int stride = gridDim.x * blockDim.x;
    int i0 = blockIdx.x * blockDim.x + threadIdx.x;
    for (int i = i0; i < n; i += stride)
        atomicAdd(&h[idx[i] & (NTYPES - 1)], 1);
    __syncthreads();
    if (threadIdx.x < NTYPES) {
        base[threadIdx.x] = atomicAdd(&cursors[threadIdx.x], h[threadIdx.x]);
        h[threadIdx.x] = 0;
    }
    __syncthreads();
    for (int i = i0; i < n; i += stride) {
        int t = idx[i] & (NTYPES - 1);
        int p = atomicAdd(&h[t], 1);
        rowlist[base[t] + p] = i;
    }
}

// Pack W[t][k][n] -> Wp[t][k/2][n][2]: a B fragment (K pair at fixed n) is
// then one aligned 8-byte load.
__global__ void il_pack_kernel(const float* __restrict__ W, float* __restrict__ Wp) {
    int i = blockIdx.x * blockDim.x + threadIdx.x;   // over 16*256*256
    int t = i >> 16;
    int k = (i >> 8) & 255;
    int n = i & 255;
    Wp[(t << 16) + ((k >> 1) << 9) + (n << 1) + (k & 1)] = W[i];
}

// ---------------- main GEMM kernel (fp32 WMMA) ----------------
//
// Workgroup = 8 wave32 waves, 64 rows x 256 cols of output.
// Wave w: output columns [32w, 32w+32) for all 4 row sub-tiles.
// acc[rt][nt]: 4 sub-tiles x 2 n-tiles x v8f = 64 VGPRs of accumulator.
// Inner loop (k += 4): 2 b64 B loads (packed W, reused 4x) + 4 ds b64 A loads
// -> 8 v_wmma_f32_16x16x4_f32.

__global__ __launch_bounds__(256) void il_gemm_kernel(
        const float* __restrict__ x,
        const float* __restrict__ Wp,
        const int* __restrict__ rowlist,
        const int* __restrict__ tileType,
        const int* __restrict__ tileRowBase,
        float* __restrict__ out,
        float* __restrict__ dummy) {
    __shared__ float ldsA[ROWS_PER_BLOCK * LDSW];
    __shared__ int srows[ROWS_PER_BLOCK];

    const int j = blockIdx.x;
    const int t = tileType[j];
    if (t < 0) return;                        // uniform whole-block exit
    const int rb  = tileRowBase[j];
    const int tid = threadIdx.x;

    if (tid < ROWS_PER_BLOCK) srows[tid] = rowlist[rb + tid];
    __syncthreads();

    // Stage A: 64 rows x 256 floats. 16 threads/row, 4 row-groups.
    {
        const int ac = (tid & 15) * 16;
        const float4 z4 = {0.f, 0.f, 0.f, 0.f};
        #pragma unroll
        for (int g = 0; g < 4; ++g) {
            const int ar  = g * 16 + (tid >> 4);
            const int row = srows[ar];
            const float* xr = x + (size_t)row * DIN;
            #pragma unroll
            for (int i = 0; i < 4; ++i) {
                float4 v = (row >= 0) ? *(const float4*)(xr + ac + 4 * i) : z4;
                *(float4*)(&ldsA[ar * LDSW + ac + 4 * i]) = v;
            }
        }
    }
    __syncthreads();

    const int wave = tid >> 5;
    const int lane = tid & 31;
    const int r    = lane & 15;               // M row (A) / N col-in-tile (B,D)
    const int kg   = lane >> 4;               // K-pair group (half-wave)
    const int n0a  = wave * 32;
    const int n0b  = n0a + 16;

    // Packed B: element {K=2p+v, N=n} at Wp[t<<16 | p<<9 | n<<1 | v].
    const float* Wpa = Wp + ((size_t)t << 16) + ((n0a + r) << 1);
    const float* Wpb = Wp + ((size_t)t << 16) + ((n0b + r) << 1);
    const float* As  = &ldsA[r * LDSW + 2 * kg];

    v8f acc[4][2];
    #pragma unroll
    for (int rt = 0; rt < 4; ++rt) {
        acc[rt][0] = (v8f){};
        acc[rt][1] = (v8f){};
    }

    #pragma unroll 4
    for (int k = 0; k < DIN; k += 4) {
        // B fragments: lane L, VGPR v -> B[K = k + 2*kg + v][N = n0 + r]
        const int koff = ((k >> 1) + kg) << 9;
        v2f b0 = *(const v2f*)(Wpa + koff);
        v2f b1 = *(const v2f*)(Wpb + koff);
        #pragma unroll
        for (int rt = 0; rt < 4; ++rt) {
            // A fragment: lane L, VGPR v -> A[M = r][K = k + 2*kg + v]
            v2f a = *(const v2f*)(As + rt * (16 * LDSW) + k);
            acc[rt][0] = __builtin_amdgcn_wmma_f32_16x16x4_f32(
                             false, a, false, b0, (short)0, acc[rt][0], false, false);
            acc[rt][1] = __builtin_amdgcn_wmma_f32_16x16x4_f32(
                             false, a, false, b1, (short)0, acc[rt][1], false, false);
        }
    }

    // D layout: VGPR i, lanes 0-15 -> M=i, lanes 16-31 -> M=i+8, N = lane%16.
    // Branchless: pad rows (rr < 0) write to a dummy scratch row (never read).
    const int half = lane >> 4;
    const int col  = lane & 15;
    #pragma unroll
    for (int rt = 0; rt < 4; ++rt) {
        #pragma unroll
        for (int i = 0; i < 8; ++i) {
            const int rr = srows[rt * 16 + 8 * half + i];
            float* o = (rr >= 0) ? (out + (size_t)rr * DOUT) : dummy;
            o[n0a + col] = acc[rt][0][i];
            o[n0b + col] = acc[rt][1][i];
        }
    }
}

// ---------------- host launcher ----------------

extern "C" void kernel_launch(void* const* d_in, const int* in_sizes, int n_in,
                              void* d_out, int out_size, void* d_ws, size_t ws_size,
                              hipStream_t stream) {
    const float* x   = (const float*)d_in[0];
    const int*   idx = (const int*)d_in[1];
    const float* W   = (const float*)d_in[2];
    float*       out = (float*)d_out;
    const int N = in_sizes[1];
    const int maxTiles = N / ROWS_PER_BLOCK + NTYPES;   // worst-case supertiles

    // Workspace layout (ints):
    //  [0..15]   counts
    //  [32..47]  cursors
    //  [64 ..)   tileType[maxTiles], tileRowBase[maxTiles],
    //            rowlist[N + 16*ROWS_PER_BLOCK], Wp[16*256*256 floats],
    //            dummy[256 floats]
    int* ws          = (int*)d_ws;
    int* counts      = ws;
    int* cursors     = ws + 32;
    int* tileType    = ws + 64;
    int* tileRowBase = tileType + maxTiles;
    int* rowlist     = tileRowBase + maxTiles;
    float* Wp        = (float*)(rowlist + N + NTYPES * ROWS_PER_BLOCK);
    float* dummy     = Wp + (size_t)NTYPES * DIN * DOUT;

    il_init_kernel<<<1, 64, 0, stream>>>(counts);
    il_hist_kernel<<<256, 256, 0, stream>>>(idx, counts, N);
    il_setup_kernel<<<1, 256, 0, stream>>>(counts, cursors, tileType,
                                           tileRowBase, rowlist, maxTiles);
    il_scatter_kernel<<<256, 256, 0, stream>>>(idx, cursors, rowlist, N);
    il_pack_kernel<<<(NTYPES * DIN * DOUT) / 256, 256, 0, stream>>>(W, Wp);
    il_gemm_kernel<<<maxTiles, 256, 0, stream>>>(x, Wp, rowlist, tileType,
                                                 tileRowBase, out, dummy);
}